// NX_CDRModel_40020505264238
// MI455X (gfx1250) — compile-verified
//
#include <hip/hip_runtime.h>
#include <hip/hip_bf16.h>

// ---------------------------------------------------------------------------
// Problem constants (must match the reference)
// ---------------------------------------------------------------------------
#define BATCH   4096
#define NTEXT   8192
#define NPOS    8
#define KNEG    1016                 // TBS - P = 1024 - 8
#define PADCNT  3079.0f              // B - (1 + K) zero-padded logits -> e^0 each
#define A_C     1.57694f
#define BB_C    0.89506f
#define EPS_C   1e-12f
#define SCALE_C 9.6179669392592631f  // (1/TEMP) * log2(e),  TEMP = 0.15
#define INVT_C  6.6666666666666667f  // 1/TEMP
#define LN2_C   0.6931471805599453f
#define ROWS_PER_WG 16
#define BLOCK   256                  // 8 wave32s
#define MASKW   (NTEXT / 32)         // 256 u32 words per row bitmask
#define KPT     (NTEXT / BLOCK)      // 32 keys per thread (stride BLOCK)

typedef __attribute__((ext_vector_type(2))) float v2f;
typedef __attribute__((ext_vector_type(8))) float v8f;

// ---------------------------------------------------------------------------
// Counter-based RNG (Threefry-2x32, 20 rounds) standing in for jax fold_in key
// ---------------------------------------------------------------------------
__device__ __forceinline__ void threefry2x32(unsigned x0, unsigned x1,
                                             unsigned& o0, unsigned& o1) {
  const unsigned k0 = 0x6C078965u, k1 = 0x9E3779B9u;   // "folded" key consts
  const unsigned ks2 = k0 ^ k1 ^ 0x1BD11BDAu;
  x0 += k0; x1 += k1;
#define TFR(r) { x0 += x1; x1 = (x1 << (r)) | (x1 >> (32 - (r))); x1 ^= x0; }
  TFR(13) TFR(15) TFR(26) TFR(6)
  x0 += k1;  x1 += ks2 + 1u;
  TFR(17) TFR(29) TFR(16) TFR(24)
  x0 += ks2; x1 += k0 + 2u;
  TFR(13) TFR(15) TFR(26) TFR(6)
  x0 += k0;  x1 += k1 + 3u;
  TFR(17) TFR(29) TFR(16) TFR(24)
  x0 += k1;  x1 += ks2 + 4u;
  TFR(13) TFR(15) TFR(26) TFR(6)
  x0 += ks2; x1 += k0 + 5u;
#undef TFR
  o0 = x0; o1 = x1;
}

// order-preserving float -> uint mapping (larger float => larger key)
__device__ __forceinline__ unsigned fkey(float f) {
  unsigned b = __float_as_uint(f);
  return (b & 0x80000000u) ? ~b : (b | 0x80000000u);
}

// Gumbel-perturbed log-weight score key for element (row, col)
__device__ __forceinline__ unsigned gumbel_key(int row, int col, float logw) {
  unsigned o0, o1;
  threefry2x32((unsigned)row, (unsigned)col, o0, o1);
  float u  = fmaxf((float)(o0 >> 8) * 5.9604644775390625e-08f, 1e-20f);
  float l1 = __builtin_amdgcn_logf(u) * LN2_C;          // ln(u) < 0
  float g  = -(__builtin_amdgcn_logf(-l1) * LN2_C);     // -ln(-ln(u))
  return fkey(logw + g);
}

// UMAP similarity from dot product and squared norms (native v_log/v_exp/v_rcp)
__device__ __forceinline__ float umap_sim(float dot, float xs, float ts) {
  float d2 = fmaxf(xs + ts - 2.0f * dot, EPS_C);
  float p  = __builtin_amdgcn_exp2f(BB_C * __builtin_amdgcn_logf(d2));
  return __builtin_amdgcn_rcpf(1.0f + A_C * p);
}

__global__ void zero_out_kernel(float* out) { out[0] = 0.0f; }

// one 16-col tile: WMMA dot products + branchless masked UMAP/exp accumulation
__device__ __forceinline__ void tile_accum(int t, int lane, int mr, int mhi,
                                           const v2f& a,
                                           const float* __restrict__ sText,
                                           const float* __restrict__ sTs,
                                           const unsigned* __restrict__ sMask,
                                           const float xsr[8],
                                           float* acc) {
  const int n = t * 16 + mr;
  // unconditional ds_load_b64 + cndmask (no exec juggling)
  v2f tv = *(const v2f*)(sText + 2 * n);
  v2f b;
  b.x = (lane < 16) ? tv.x : 0.0f;
  b.y = (lane < 16) ? tv.y : 0.0f;

  v8f c = {};
  c = __builtin_amdgcn_wmma_f32_16x16x4_f32(false, a, false, b,
                                            (short)0, c, false, false);
  const float tsn = sTs[n];
  const int   w   = n >> 5;
  const int   bit = n & 31;
  // 8 mask words (m = mhi..mhi+7) as two ds_load_b128
  uint4 ma = *(const uint4*)(sMask + w * 16 + mhi);
  uint4 mb = *(const uint4*)(sMask + w * 16 + mhi + 4);
  unsigned mw[8] = {ma.x, ma.y, ma.z, ma.w, mb.x, mb.y, mb.z, mb.w};
  // branchless: take-rate per v across a wave is ~99%, so exec-branching only
  // adds overhead; cndmask-select lets the 8 trans chains interleave
#pragma unroll
  for (int v = 0; v < 8; ++v) {
    float d2 = fmaxf(xsr[v] + tsn - 2.0f * c[v], EPS_C);
    float pw = __builtin_amdgcn_exp2f(BB_C * __builtin_amdgcn_logf(d2));
    float s  = __builtin_amdgcn_rcpf(1.0f + A_C * pw);
    float e  = __builtin_amdgcn_exp2f(s * SCALE_C);    // exp(sim / TEMP)
    acc[v]  += ((mw[v] >> bit) & 1u) ? e : 0.0f;
  }
}

// ---------------------------------------------------------------------------
// Fused kernel: 16 rows per workgroup.
//  Phase 0: stage text embeddings, |t|^2, 0.75*log(wf) into LDS
//  Phase 1: per row, keys live in registers (32/thread, stride-256); exact
//           Kth-largest via 4-pass byte radix histogram; selection bitmask
//           baked with wave32 __ballot into word-major LDS mask
//  Phase 2: sweep 512 column tiles (2x unrolled) with v_wmma_f32_16x16x4_f32;
//           branchless epilogue: bitmask select + UMAP kernel + exp accum
// ---------------------------------------------------------------------------
__global__ __launch_bounds__(BLOCK)
void cdr_loss_kernel(const float* __restrict__ xEmb,
                     const float* __restrict__ tEmb,
                     const float* __restrict__ wf,
                     const int*   __restrict__ posIdx,
                     float* __restrict__ out) {
  extern __shared__ unsigned char smem_raw[];
  float*    sText = (float*)smem_raw;             // NTEXT*2
  float*    sLogw = sText + NTEXT * 2;            // NTEXT
  float*    sTs   = sLogw + NTEXT;                // NTEXT
  unsigned* sMask = (unsigned*)(sTs + NTEXT);     // MASKW*16, layout [w*16+m]
  unsigned* sHist = sMask + MASKW * 16;           // 256 bins
  unsigned* sSel  = sHist + 256;                  // {bin, krem}
  float*    sX    = (float*)(sSel + 2);           // 16*2
  float*    sXs   = sX + 32;                      // 16
  int*      sPos  = (int*)(sXs + 16);             // 16*8
  float*    sPsum = (float*)(sPos + 128);         // 16
  float*    sNsum = sPsum + 16;                   // 16

  const int tid  = threadIdx.x;
  const int row0 = blockIdx.x * ROWS_PER_WG;
  const int wave = tid >> 5;
  const int lane = tid & 31;

  __builtin_prefetch(tEmb, 0, 3);                 // global_prefetch_b8

  // ---- Phase 0: staging --------------------------------------------------
  for (int j = tid; j < NTEXT; j += BLOCK) {
    float ex = tEmb[2 * j + 0];
    float ey = tEmb[2 * j + 1];
    sText[2 * j + 0] = ex;
    sText[2 * j + 1] = ey;
    sTs[j]   = ex * ex + ey * ey;
    sLogw[j] = 0.75f * (__builtin_amdgcn_logf(wf[j]) * LN2_C);
  }
  if (tid < 16) {
    float ex = xEmb[2 * (row0 + tid) + 0];
    float ey = xEmb[2 * (row0 + tid) + 1];
    sX[2 * tid + 0] = ex;
    sX[2 * tid + 1] = ey;
    sXs[tid]  = ex * ex + ey * ey;
    sPsum[tid] = 0.0f;
    sNsum[tid] = 0.0f;
  }
  if (tid < 128) sPos[tid] = posIdx[(row0 + (tid >> 3)) * NPOS + (tid & 7)];
  __syncthreads();

  // positives: 128 threads, one per (row, p)
  if (tid < 128) {
    int m = tid >> 3;
    int j = sPos[tid];
    float s = umap_sim(sX[2 * m] * sText[2 * j] + sX[2 * m + 1] * sText[2 * j + 1],
                       sXs[m], sTs[j]);
    atomicAdd(&sPsum[m], s * 0.125f);
  }
  __syncthreads();

  // ---- Phase 1: register-resident keys + radix select + ballot mask ------
  unsigned keys[KPT];
#pragma unroll 1
  for (int m = 0; m < ROWS_PER_WG; ++m) {
    const int rowg = row0 + m;
    // fill: thread owns j = tid + 256*k  (conflict-free sLogw reads)
#pragma unroll 1
    for (int k = 0; k < KPT; ++k) {
      const int j = tid + BLOCK * k;
      bool isPos = false;
#pragma unroll
      for (int p = 0; p < NPOS; ++p) isPos |= (sPos[m * NPOS + p] == j);
      keys[k] = isPos ? 0u : gumbel_key(rowg, j, sLogw[j]);
    }

    // 4-pass byte-radix select for the Kth largest key
    unsigned prefix = 0u;
    int      krem   = KNEG;
#pragma unroll 1
    for (int p = 0; p < 4; ++p) {
      const int      shift  = 24 - 8 * p;
      const unsigned himask = (p == 0) ? 0u : (0xFFFFFFFFu << (32 - 8 * p));
      sHist[tid] = 0u;
      __syncthreads();
#pragma unroll
      for (int k = 0; k < KPT; ++k) {
        if ((keys[k] & himask) == prefix)
          atomicAdd(&sHist[(keys[k] >> shift) & 0xFFu], 1u);
      }
      __syncthreads();
      if (tid == 0) {
        unsigned cum = 0u;
        int b = 255;
        for (; b > 0; --b) {
          cum += sHist[b];
          if ((int)cum >= krem) break;
        }
        if ((int)cum < krem) cum += sHist[0];   // b == 0 fallthrough
        sSel[0] = (unsigned)b;
        sSel[1] = (unsigned)(krem - (int)(cum - sHist[b]));
      }
      __syncthreads();
      prefix |= sSel[0] << shift;
      krem    = (int)sSel[1];
    }

    // bake selection mask via wave32 ballot: lanes of (wave,k) cover word w
#pragma unroll
    for (int k = 0; k < KPT; ++k) {
      unsigned word = (unsigned)__ballot(keys[k] >= prefix);
      if (lane == 0) sMask[(wave + 8 * k) * 16 + m] = word;
    }
    __syncthreads();
  }

  // ---- Phase 2: WMMA sweep over 512 column tiles (2x unrolled) -----------
  const int mr  = lane & 15;
  const int mhi = (lane >> 4) << 3;    // 0 for lanes 0-15, 8 for lanes 16-31

  // A matrix (16x4 f32): lanes 0-15 carry K=0,1 ; lanes 16-31 carry K=2,3 (zero)
  v2f xv = *(const v2f*)(sX + 2 * mr);
  v2f a;
  a.x = (lane < 16) ? xv.x : 0.0f;
  a.y = (lane < 16) ? xv.y : 0.0f;

  // hoist the 8 loop-invariant |x_m|^2 values into registers
  float xsr[8];
#pragma unroll
  for (int v = 0; v < 8; ++v) xsr[v] = sXs[v + mhi];

  float acc[8];
#pragma unroll
  for (int v = 0; v < 8; ++v) acc[v] = 0.0f;

#pragma unroll 1
  for (int t = wave; t < NTEXT / 16; t += 16) {  // uniform per wave, EXEC all-1s
    tile_accum(t,     lane, mr, mhi, a, sText, sTs, sMask, xsr, acc);
    tile_accum(t + 8, lane, mr, mhi, a, sText, sTs, sMask, xsr, acc);
  }

#pragma unroll
  for (int v = 0; v < 8; ++v) atomicAdd(&sNsum[v + mhi], acc[v]);
  __syncthreads();

  // ---- Final: per-row loss, reduce to scalar -----------------------------
  if (tid < 16) {
    float z0    = sPsum[tid] * INVT_C;
    float total = __builtin_amdgcn_exp2f(z0 * 1.4426950408889634f)
                  + sNsum[tid] + PADCNT;
    float li    = __builtin_amdgcn_logf(total) * LN2_C - z0;
    atomicAdd(out, li * (1.0f / (float)BATCH));
  }
}

// ---------------------------------------------------------------------------
extern "C" void kernel_launch(void* const* d_in, const int* in_sizes, int n_in,
                              void* d_out, int out_size, void* d_ws, size_t ws_size,
                              hipStream_t stream) {
  const float* xEmb   = (const float*)d_in[0];   // (B, 2)  f32
  const float* tEmb   = (const float*)d_in[1];   // (NT, 2) f32
  const float* wf     = (const float*)d_in[2];   // (NT,)   f32
  const int*   posIdx = (const int*)  d_in[3];   // (B, P)  i32
  float*       out    = (float*)d_out;           // scalar  f32

  zero_out_kernel<<<1, 1, 0, stream>>>(out);

  const size_t smem =
      (size_t)NTEXT * 2 * sizeof(float) +        // sText
      (size_t)NTEXT * sizeof(float) +            // sLogw
      (size_t)NTEXT * sizeof(float) +            // sTs
      (size_t)MASKW * 16 * sizeof(unsigned) +    // sMask
      256 * sizeof(unsigned) +                   // sHist
      2 * sizeof(unsigned) +                     // sSel
      (32 + 16) * sizeof(float) +                // sX, sXs
      128 * sizeof(int) +                        // sPos
      (16 + 16) * sizeof(float);                 // sPsum, sNsum

  cdr_loss_kernel<<<BATCH / ROWS_PER_WG, BLOCK, smem, stream>>>(
      xEmb, tEmb, wf, posIdx, out);
}